// TTConv_37374805410123
// MI455X (gfx1250) — compile-verified
//
#include <hip/hip_runtime.h>
#include <stdint.h>

// ---------------------------------------------------------------------------
// TT-conv + Cayley for MI455X (gfx1250, wave32).
//   einsum : y[o,b,k,m,j] = sum_i W[i,m,o] * c'[i,b,k,m,j]  via V_WMMA_F32_16X16X4_F32
//   cayley : Q = (I-A)^-1 (I+A) per (o,b,m), A 32x32 skew; TDM tile load to LDS,
//            Gauss-Jordan, in-place on the d_out slice (L2-resident round trip).
// All shapes templated -> branchless unrolled inner loops, 32-bit addressing.
// ---------------------------------------------------------------------------

typedef __attribute__((ext_vector_type(2))) float v2f;
typedef __attribute__((ext_vector_type(8))) float v8f;
typedef __attribute__((ext_vector_type(4))) unsigned int u32x4;
typedef __attribute__((ext_vector_type(4))) int i32x4;
typedef __attribute__((ext_vector_type(8))) int i32x8;

#define I_CH  32
#define O_CH  128
#define BATCH 64
#define JJ    24   // 4+8+8+4 rows of weight

// ---------------------------------------------------------------------------
// GEMM: one wave -> one 16(o) x 16(n) f32 tile, n = flat (b,k,j) for fixed m.
// i-dim (32) consumed in 8 WMMA steps. ANTISYM fuses c' = 0.5*(c[k<->j] - c).
// ---------------------------------------------------------------------------
template<int K, int MDIM, int J, bool ANTISYM>
__global__ __launch_bounds__(128) void tt_gemm(
    const float* __restrict__ core,    // (I, B, K, M, J)
    const float* __restrict__ weight,  // (I, 24, 128)
    float* __restrict__ out,           // (O, B, K, M, J)
    int m_off)
{
    constexpr int ISTRIDE = BATCH * K * MDIM * J;   // i stride == o stride (elements)

    const int lane = threadIdx.x & 31;
    const int wave = threadIdx.x >> 5;
    const int m    = blockIdx.z;
    const int o0   = blockIdx.y * 16;
    const int n0   = (blockIdx.x * 4 + wave) * 16;

    const int half = lane >> 4;        // 0: lanes 0-15, 1: lanes 16-31
    const int l16  = lane & 15;

    // decode this lane's column n -> (b,k,j); all divisors compile-time
    const int n  = n0 + l16;
    const int j  = n % J;
    const int kk = (n / J) % K;
    const int b  = n / (J * K);

    const int base = ((b * K + kk) * MDIM + m) * J + j;     // element @ i=0
    int swapbase = base;
    if constexpr (ANTISYM) swapbase = ((b * K + j) * MDIM + m) * J + kk;

    // A operand = W^T panel: a.x/a.y = W[i_sel(+1), m_off+m, o0+l16]
    const float* wbase = weight + (m_off + m) * O_CH + (o0 + l16);
    __builtin_prefetch(wbase, 0, 1);   // global_prefetch_b8 of the W panel

    v8f acc = {};
    #pragma unroll
    for (int i0 = 0; i0 < I_CH; i0 += 4) {
        const int i_sel = i0 + 2 * half;     // VGPR0 <- i_sel, VGPR1 <- i_sel+1
        v2f a, bb;
        a.x = wbase[(i_sel    ) * (JJ * O_CH)];
        a.y = wbase[(i_sel + 1) * (JJ * O_CH)];
        if constexpr (ANTISYM) {
            const float c0 = core[(i_sel    ) * ISTRIDE + base];
            const float s0 = core[(i_sel    ) * ISTRIDE + swapbase];
            const float c1 = core[(i_sel + 1) * ISTRIDE + base];
            const float s1 = core[(i_sel + 1) * ISTRIDE + swapbase];
            bb.x = 0.5f * (s0 - c0);
            bb.y = 0.5f * (s1 - c1);
        } else {
            bb.x = core[(i_sel    ) * ISTRIDE + base];
            bb.y = core[(i_sel + 1) * ISTRIDE + base];
        }
        acc = __builtin_amdgcn_wmma_f32_16x16x4_f32(
                  false, a, false, bb, (short)0, acc, false, false);
    }

    // C layout: VGPR r holds row o = o0 + r + 8*half, col = n
    float* obase = out + base;
    #pragma unroll
    for (int r = 0; r < 8; ++r)
        obase[(o0 + r + 8 * half) * ISTRIDE] = acc[r];
}

// ---------------------------------------------------------------------------
// Batched Cayley. One block per (o,b,m) matrix. The 32x32 tile (row stride
// 256 elems, rows contiguous 32 elems) is fetched by the Tensor Data Mover
// into LDS (pitch 33 dwords via D# padding: 1 dword pad each 32 dwords),
// then Gauss-Jordan on a 32x66 augmented tile (stride 66 -> conflict-free
// column reads on 64 banks). In-place: blocks own disjoint matrices.
// ---------------------------------------------------------------------------
#if defined(__has_builtin) && __has_builtin(__builtin_amdgcn_tensor_load_to_lds)
  #define USE_TDM 1
  #if __has_include(<hip/amd_detail/amd_gfx1250_TDM.h>)
    #define TDM_6ARG 1      // amdgpu-toolchain (clang-23/therock): 6-arg builtin
  #else
    #define TDM_6ARG 0      // ROCm 7.2 (clang-22): 5-arg builtin
  #endif
#else
  #define USE_TDM 0
#endif

#define LDA  66    // augmented tile pitch (dwords)
#define SPW  33    // TDM staging pitch (dwords): 32 data + 1 pad

template<int MDIM>
__global__ __launch_bounds__(256) void tt_cayley(float* __restrict__ y)
{
    __shared__ float As[32 * SPW];     // raw A, staged by TDM
    __shared__ float aug[32 * LDA];    // [ I-A | I+A ]

    const int idx = blockIdx.x;
    const int m = idx % MDIM;
    const int b = (idx / MDIM) % BATCH;
    const int o = idx / (MDIM * BATCH);
    constexpr int KSTRIDE = MDIM * 32;                         // k stride (J=32)
    const int base = (o * BATCH + b) * 32 * KSTRIDE + m * 32;

    const int t = threadIdx.x;

#if USE_TDM
    if (t < 32) {   // wave 0 issues the tensor DMA (EXEC ignored by TDM)
        const unsigned long long ga = (unsigned long long)(uintptr_t)(y + base);
        const unsigned lds = (unsigned)(uintptr_t)(void*)As;
        u32x4 g0;
        g0[0] = 1u;                                            // count=1 valid D#
        g0[1] = lds;                                           // lds_addr (bytes)
        g0[2] = (unsigned)(ga & 0xFFFFFFFFu);                  // global_addr[31:0]
        g0[3] = (unsigned)((ga >> 32) & 0x1FFFFFFu) | (2u << 30); // [56:32], type=2
        i32x8 g1;
        g1[0] = (2 << 16)     // data_size = 4B
              | (1 << 20)     // pad_enable
              | (4 << 22)     // pad_interval: every 32 dwords
              | (0 << 25);    // pad_amount: 1 dword  -> LDS pitch 33 dwords
        g1[1] = (32 << 16);                 // tensor_dim0[15:0] = 32
        g1[2] = (32 << 16);                 // dim0[31:16]=0 | tensor_dim1[15:0]=32
        g1[3] = (32 << 16);                 // dim1[31:16]=0 | tile_dim0 = 32
        g1[4] = 32;                         // tile_dim1 = 32 | tile_dim2 = 0
        g1[5] = KSTRIDE;                    // tensor_dim0_stride[31:0] = 256
        g1[6] = 0;                          // stride0[47:32] | stride1[15:0]
        g1[7] = 0;
        const i32x4 gz = {0, 0, 0, 0};      // groups 2/3 unused (2D tensor)
#if TDM_6ARG
        const i32x8 gz8 = {0, 0, 0, 0, 0, 0, 0, 0};
        __builtin_amdgcn_tensor_load_to_lds(g0, g1, gz, gz, gz8, 0);
#else
        __builtin_amdgcn_tensor_load_to_lds(g0, g1, gz, gz, 0);
#endif
        __builtin_amdgcn_s_wait_tensorcnt(0);                  // s_wait_tensorcnt 0
    }
    __syncthreads();
    #pragma unroll
    for (int q = 0; q < 4; ++q) {
        const int e = t + 256 * q;
        const int k = e >> 5, jj = e & 31;
        const float a  = As[k * SPW + jj];
        const float id = (k == jj) ? 1.0f : 0.0f;
        aug[k * LDA + jj]      = id - a;
        aug[k * LDA + 32 + jj] = id + a;
    }
#else
    #pragma unroll
    for (int q = 0; q < 4; ++q) {
        const int e = t + 256 * q;
        const int k = e >> 5, jj = e & 31;
        const float a  = y[base + k * KSTRIDE + jj];
        const float id = (k == jj) ? 1.0f : 0.0f;
        aug[k * LDA + jj]      = id - a;
        aug[k * LDA + 32 + jj] = id + a;
    }
#endif
    __syncthreads();

    const int c  = t & 63;   // my column of the augmented system
    const int kb = t >> 6;   // my row group; rows k = kb + 4*s
    #pragma unroll 1
    for (int r = 0; r < 32; ++r) {
        const float prow = aug[r * LDA + c];
        const float invp = 1.0f / aug[r * LDA + r];
        float f[8];
        #pragma unroll
        for (int s = 0; s < 8; ++s)
            f[s] = aug[(kb + 4 * s) * LDA + r] * invp;
        __syncthreads();
        #pragma unroll
        for (int s = 0; s < 8; ++s) {
            const int k = kb + 4 * s;
            if (k == r) aug[k * LDA + c]  = prow * invp;   // normalize pivot row
            else        aug[k * LDA + c] -= f[s] * prow;   // eliminate
        }
        __syncthreads();
    }

    // right half now holds Q = (I-A)^-1 (I+A)
    #pragma unroll
    for (int q = 0; q < 4; ++q) {
        const int e = t + 256 * q;
        const int k = e >> 5, jj = e & 31;
        y[base + k * KSTRIDE + jj] = aug[k * LDA + 32 + jj];
    }
}

extern "C" void kernel_launch(void* const* d_in, const int* in_sizes, int n_in,
                              void* d_out, int out_size, void* d_ws, size_t ws_size,
                              hipStream_t stream) {
    const float* core0  = (const float*)d_in[0];
    const float* core1  = (const float*)d_in[1];
    const float* core2  = (const float*)d_in[2];
    const float* core3  = (const float*)d_in[3];
    const float* weight = (const float*)d_in[4];
    float* out = (float*)d_out;

    float* out0 = out;                 // (128,64, 1,4,32) = 1,048,576
    float* out1 = out0 + 1048576;      // (128,64,32,8,32) = 8,388,608
    float* out2 = out1 + 8388608;      // (128,64,32,8,32) = 8,388,608
    float* out3 = out2 + 8388608;      // (128,64,32,4, 1) = 1,048,576

    // core0: K=1,M=4,J=32 -> N=2048 per m (128 n-tiles / 4 waves = 32 blocks)
    tt_gemm< 1, 4, 32, false><<<dim3(32,   8, 4), 128, 0, stream>>>(core0, weight, out0,  0);
    // core1: antisym GEMM -> A into out1, in-place Cayley (L2-resident)
    tt_gemm<32, 8, 32, true ><<<dim3(1024, 8, 8), 128, 0, stream>>>(core1, weight, out1,  4);
    tt_cayley<8><<<dim3(128 * 64 * 8), 256, 0, stream>>>(out1);
    // core2
    tt_gemm<32, 8, 32, true ><<<dim3(1024, 8, 8), 128, 0, stream>>>(core2, weight, out2, 12);
    tt_cayley<8><<<dim3(128 * 64 * 8), 256, 0, stream>>>(out2);
    // core3: K=32,M=4,J=1 -> N=2048 per m
    tt_gemm<32, 4,  1, false><<<dim3(32,   8, 4), 128, 0, stream>>>(core3, weight, out3, 20);
}